// SimplifiedPrototypeGNN_37297495998545
// MI455X (gfx1250) — compile-verified
//
#include <hip/hip_runtime.h>
#include <hip/hip_bf16.h>
#include <stdint.h>

#define Nn 8192
#define Dd 128
#define Hh 2
#define Kk 3
#define Ee (Nn*Kk)

typedef float v2f __attribute__((ext_vector_type(2)));
typedef float v4f __attribute__((ext_vector_type(4)));
typedef float v8f __attribute__((ext_vector_type(8)));

// LDS tile layout: 16 rows x 132 floats (128 data + 4 pad) -> conflict-free b64 reads
#define TROW 132
#define TBUF (16 * TROW)   // floats per buffer (2112), 8448 bytes

typedef __attribute__((address_space(3))) char as3_char;
__device__ __forceinline__ unsigned lds_byte_off(void* p) {
  return (unsigned)(unsigned long long)(as3_char*)p;   // addrspacecast -> LDS byte offset
}

// CDNA5 async copy: 16B per lane, global -> LDS (tracked by ASYNCcnt)
__device__ __forceinline__ void async_copy16(unsigned lds_off, const void* gptr) {
  asm volatile("global_load_async_to_lds_b128 %0, %1, off"
               :: "v"(lds_off), "v"(gptr) : "memory");
}
__device__ __forceinline__ void wait_async0() {
  asm volatile("s_wait_asynccnt 0" ::: "memory");
}

// ---------------- row squared norms ----------------
__global__ __launch_bounds__(256) void k_sq(const float* __restrict__ x, float* __restrict__ sq) {
  int r = blockIdx.x * 256 + threadIdx.x;
  if (r >= Nn) return;
  const v4f* p = (const v4f*)(x + r * Dd);
  float s = 0.f;
#pragma unroll
  for (int i = 0; i < Dd / 4; ++i) { v4f a = p[i]; s += a.x*a.x + a.y*a.y + a.z*a.z + a.w*a.w; }
  sq[r] = s;
}

// ---------------- kNN top-3 via fp32 WMMA Gram tiles, async double-buffered LDS ----------------
// Wave handles 16 query rows (N dim of D). Block = 4 waves = 64 rows. Grid = 128.
__global__ __launch_bounds__(128) void k_knn(const float* __restrict__ x,
                                             const float* __restrict__ sq,
                                             int* __restrict__ knn) {
  __shared__ __align__(16) float lds[2 * TBUF];

  const int tid  = threadIdx.x;
  const int lane = tid & 31;
  const int wave = tid >> 5;
  const int m    = lane & 15;
  const int hi   = lane >> 4;                 // half-wave (K / M split)
  const int Q    = blockIdx.x * 64 + wave * 16;

  // Resident query fragments (B operand): q[t] covers K = 4t..4t+3 (lo lanes K+0,1; hi lanes K+2,3)
  v2f q[32];
  {
    const float* qp = x + (Q + m) * Dd + 2 * hi;
#pragma unroll
    for (int t = 0; t < 32; ++t) q[t] = *(const v2f*)(qp + 4 * t);
  }

  const unsigned lbase = lds_byte_off(lds);
  // issue async copy of candidate tile 0 into buffer 0 (each thread copies 4x16B of the 8KB tile)
  {
    const char* g = (const char*)x;           // tile 0 is x[0..15][*], contiguous 8KB
#pragma unroll
    for (int i = 0; i < 4; ++i) {
      int ci = tid + i * 128;                 // 16B chunk index in tile (0..511)
      unsigned lo = lbase + (unsigned)((ci >> 5) * (TROW * 4) + (ci & 31) * 16);
      async_copy16(lo, g + ci * 16);
    }
  }

  float d0 = 3.4e38f, d1 = 3.4e38f, d2v = 3.4e38f;
  int   i0 = 0,       i1 = 0,       i2 = 0;

  for (int c = 0; c < Nn / 16; ++c) {
    wait_async0();          // this wave's copies for tile c are complete
    __syncthreads();        // all waves' copies visible; all waves done reading the other buffer

    if (c + 1 < Nn / 16) {  // prefetch next tile into the other buffer
      const char* g = (const char*)x + (size_t)(c + 1) * 8192;
      unsigned bb = lbase + (unsigned)(((c + 1) & 1) * (TBUF * 4));
#pragma unroll
      for (int i = 0; i < 4; ++i) {
        int ci = tid + i * 128;
        async_copy16(bb + (unsigned)((ci >> 5) * (TROW * 4) + (ci & 31) * 16), g + ci * 16);
      }
    }

    const int cbase = c * 16;
    const float* buf = &lds[(c & 1) * TBUF];
    v8f acc = {};
#pragma unroll
    for (int t = 0; t < 32; ++t) {
      v2f a = *(const v2f*)(buf + m * TROW + 4 * t + 2 * hi);   // A operand: candidate rows
      acc = __builtin_amdgcn_wmma_f32_16x16x4_f32(false, a, false, q[t],
                                                  (short)0, acc, false, false);
    }

    // ordering key: sq[col] - 2*dot  (per-row constant sq[row] dropped)
    const float* sp = sq + cbase + 8 * hi;
    v4f s0 = *(const v4f*)sp;
    v4f s1 = *(const v4f*)(sp + 4);
    float sv[8] = {s0.x, s0.y, s0.z, s0.w, s1.x, s1.y, s1.z, s1.w};
#pragma unroll
    for (int v = 0; v < 8; ++v) {
      float key = __builtin_fmaf(-2.0f, acc[v], sv[v]);
      int   idx = cbase + 8 * hi + v;                  // ascending within lane -> stable ties
      if (key < d2v) {
        if (key < d1) {
          d2v = d1; i2 = i1;
          if (key < d0) { d1 = d0; i1 = i0; d0 = key; i0 = idx; }
          else          { d1 = key; i1 = idx; }
        } else { d2v = key; i2 = idx; }
      }
    }
  }

  // merge lane n (cols 0..7 mod 16) with lane n+16 (cols 8..15 mod 16)
  float p0 = __shfl_xor(d0, 16, 32);
  float p1 = __shfl_xor(d1, 16, 32);
  float p2 = __shfl_xor(d2v, 16, 32);
  int   j0 = __shfl_xor(i0, 16, 32);
  int   j1 = __shfl_xor(i1, 16, 32);
  int   j2 = __shfl_xor(i2, 16, 32);
  if (hi == 0) {
    float a[3] = {d0, d1, d2v}; int ai[3] = {i0, i1, i2};
    float b[3] = {p0, p1, p2};  int bi[3] = {j0, j1, j2};
    int ia = 0, ib = 0;
    int row = Q + m;
#pragma unroll
    for (int k = 0; k < 3; ++k) {
      bool takeA = (a[ia] < b[ib]) || ((a[ia] == b[ib]) && (ai[ia] < bi[ib]));
      knn[row * 3 + k] = takeA ? ai[ia] : bi[ib];
      ia += takeA ? 1 : 0;
      ib += takeA ? 0 : 1;
    }
  }
}

// ---------------- xp = x @ W via fp32 WMMA (M=8192, N=256, K=128) ----------------
__global__ __launch_bounds__(128) void k_xp(const float* __restrict__ x,
                                            const float* __restrict__ W,
                                            float* __restrict__ xp) {
  const int lane = threadIdx.x & 31;
  const int wave = threadIdx.x >> 5;
  const int tile = blockIdx.x * 4 + wave;
  const int tm   = tile >> 4;     // row block (0..511)
  const int tn   = tile & 15;     // col block (0..15)
  const int m    = lane & 15;
  const int hi   = lane >> 4;

  const float* ap = x + (tm * 16 + m) * Dd + 2 * hi;
  const float* wb = W + tn * 16 + m;           // W[k][col], row stride 256
  v8f acc = {};
#pragma unroll
  for (int t = 0; t < 32; ++t) {
    v2f a = *(const v2f*)(ap + 4 * t);
    int k0 = 4 * t + 2 * hi;
    v2f b;
    b.x = wb[(k0    ) * (Hh * Dd)];
    b.y = wb[(k0 + 1) * (Hh * Dd)];
    acc = __builtin_amdgcn_wmma_f32_16x16x4_f32(false, a, false, b,
                                                (short)0, acc, false, false);
  }
#pragma unroll
  for (int v = 0; v < 8; ++v)
    xp[(tm * 16 + v + 8 * hi) * (Hh * Dd) + tn * 16 + m] = acc[v];
}

// ---------------- attention scalars a_s, a_d ----------------
__global__ __launch_bounds__(256) void k_att(const float* __restrict__ xp,
                                             const float* __restrict__ att_src,
                                             const float* __restrict__ att_dst,
                                             float* __restrict__ a_s, float* __restrict__ a_d) {
  int idx = blockIdx.x * 256 + threadIdx.x;    // n*2 + h, < 16384
  if (idx >= Nn * Hh) return;
  int n = idx >> 1, h = idx & 1;
  const v4f* xr = (const v4f*)(xp + n * (Hh * Dd) + h * Dd);
  const v4f* as = (const v4f*)(att_src + h * Dd);
  const v4f* ad = (const v4f*)(att_dst + h * Dd);
  float ss = 0.f, sd = 0.f;
#pragma unroll
  for (int i = 0; i < Dd / 4; ++i) {
    v4f xv = xr[i], s = as[i], t = ad[i];
    ss += xv.x*s.x + xv.y*s.y + xv.z*s.z + xv.w*s.w;
    sd += xv.x*t.x + xv.y*t.y + xv.z*t.z + xv.w*t.w;
  }
  a_s[idx] = ss; a_d[idx] = sd;
}

// ---------------- zero / init softmax state ----------------
__global__ __launch_bounds__(256) void k_zero(float* __restrict__ acc,
                                              float* __restrict__ denom,
                                              int* __restrict__ m_enc) {
  int i = blockIdx.x * 256 + threadIdx.x;
  if (i < Nn * Hh * Dd) acc[i] = 0.f;
  if (i < Nn * Hh) { denom[i] = 0.f; m_enc[i] = (int)0x80000000; }
}

// float <-> order-preserving signed int encoding (for integer atomicMax)
__device__ __forceinline__ int   encf(float x) { int u = __float_as_int(x); return u >= 0 ? u : (u ^ 0x7FFFFFFF); }
__device__ __forceinline__ float decf(int u)   { return __int_as_float(u >= 0 ? u : (u ^ 0x7FFFFFFF)); }
__device__ __forceinline__ float lrelu(float x){ return x >= 0.f ? x : 0.2f * x; }

// ---------------- segment max over incoming edges ----------------
__global__ __launch_bounds__(256) void k_edge_max(const int* __restrict__ knn,
                                                  const float* __restrict__ a_s,
                                                  const float* __restrict__ a_d,
                                                  int* __restrict__ m_enc) {
  int idx = blockIdx.x * 256 + threadIdx.x;
  if (idx >= Ee * Hh) return;
  int e = idx >> 1, h = idx & 1;
  int src = e / Kk;
  int dst = knn[e];
  float ev = lrelu(a_s[src * Hh + h] + a_d[dst * Hh + h]);
  atomicMax(&m_enc[dst * Hh + h], encf(ev));
}

// ---------------- segment sum of exp ----------------
__global__ __launch_bounds__(256) void k_edge_sum(const int* __restrict__ knn,
                                                  const float* __restrict__ a_s,
                                                  const float* __restrict__ a_d,
                                                  const int* __restrict__ m_enc,
                                                  float* __restrict__ denom) {
  int idx = blockIdx.x * 256 + threadIdx.x;
  if (idx >= Ee * Hh) return;
  int e = idx >> 1, h = idx & 1;
  int src = e / Kk;
  int dst = knn[e];
  float ev = lrelu(a_s[src * Hh + h] + a_d[dst * Hh + h]);
  float w  = __expf(ev - decf(m_enc[dst * Hh + h]));
  atomicAdd(&denom[dst * Hh + h], w);
}

// ---------------- weighted message aggregation ----------------
__global__ __launch_bounds__(256) void k_edge_msg(const int* __restrict__ knn,
                                                  const float* __restrict__ a_s,
                                                  const float* __restrict__ a_d,
                                                  const int* __restrict__ m_enc,
                                                  const float* __restrict__ denom,
                                                  const float* __restrict__ xp,
                                                  float* __restrict__ acc) {
  int e   = blockIdx.x;
  int h   = threadIdx.x >> 7;
  int d   = threadIdx.x & 127;
  int src = e / Kk;
  int dst = knn[e];
  float ev    = lrelu(a_s[src * Hh + h] + a_d[dst * Hh + h]);
  float alpha = __expf(ev - decf(m_enc[dst * Hh + h])) / denom[dst * Hh + h];
  atomicAdd(&acc[(dst * Hh + h) * Dd + d], alpha * xp[(src * Hh + h) * Dd + d]);
}

// ---------------- head mean + bias + LayerNorm + ReLU + residual ----------------
__global__ __launch_bounds__(128) void k_final(const float* __restrict__ x,
                                               const float* __restrict__ acc,
                                               const float* __restrict__ bias,
                                               const float* __restrict__ gamma,
                                               const float* __restrict__ beta,
                                               float* __restrict__ out) {
  __shared__ float red[128];
  int row = blockIdx.x;
  int d   = threadIdx.x;
  float v = (acc[row * (Hh * Dd) + d] + acc[row * (Hh * Dd) + Dd + d]) * 0.5f + bias[d];

  red[d] = v; __syncthreads();
  for (int s = 64; s > 0; s >>= 1) { if (d < s) red[d] += red[d + s]; __syncthreads(); }
  float mu = red[0] * (1.f / Dd);
  __syncthreads();
  float c = v - mu;
  red[d] = c * c; __syncthreads();
  for (int s = 64; s > 0; s >>= 1) { if (d < s) red[d] += red[d + s]; __syncthreads(); }
  float var = red[0] * (1.f / Dd);

  float y = c * rsqrtf(var + 1e-5f) * gamma[d] + beta[d];
  y = y > 0.f ? y : 0.f;
  out[row * Dd + d] = x[row * Dd + d] + y;
}

extern "C" void kernel_launch(void* const* d_in, const int* in_sizes, int n_in,
                              void* d_out, int out_size, void* d_ws, size_t ws_size,
                              hipStream_t stream) {
  const float* x       = (const float*)d_in[0];   // [8192,128]
  // d_in[1] = labels (unused by forward)
  const float* W       = (const float*)d_in[2];   // [128,256]
  const float* att_src = (const float*)d_in[3];   // [2,128]
  const float* att_dst = (const float*)d_in[4];   // [2,128]
  const float* bias    = (const float*)d_in[5];   // [128]
  const float* gamma   = (const float*)d_in[6];   // [128]
  const float* beta    = (const float*)d_in[7];   // [128]
  float* out = (float*)d_out;

  char* ws = (char*)d_ws;                              // ~17.2 MB used
  float* sq    = (float*)(ws);                         // 32 KB
  int*   knn   = (int*)  (ws + 32768);                 // 96 KB
  float* xp    = (float*)(ws + 131072);                // 8 MB
  float* a_s   = (float*)(ws + 8519680);               // 64 KB
  float* a_d   = (float*)(ws + 8585216);               // 64 KB
  int*   m_enc = (int*)  (ws + 8650752);               // 64 KB
  float* denom = (float*)(ws + 8716288);               // 64 KB
  float* acc   = (float*)(ws + 8781824);               // 8 MB

  k_sq      <<<Nn / 256, 256, 0, stream>>>(x, sq);
  k_knn     <<<Nn / 64, 128, 0, stream>>>(x, sq, knn);
  k_xp      <<<(Nn / 16) * (Hh * Dd / 16) / 4, 128, 0, stream>>>(x, W, xp);
  k_att     <<<(Nn * Hh + 255) / 256, 256, 0, stream>>>(xp, att_src, att_dst, a_s, a_d);
  k_zero    <<<(Nn * Hh * Dd + 255) / 256, 256, 0, stream>>>(acc, denom, m_enc);
  k_edge_max<<<(Ee * Hh + 255) / 256, 256, 0, stream>>>(knn, a_s, a_d, m_enc);
  k_edge_sum<<<(Ee * Hh + 255) / 256, 256, 0, stream>>>(knn, a_s, a_d, m_enc, denom);
  k_edge_msg<<<Ee, 256, 0, stream>>>(knn, a_s, a_d, m_enc, denom, xp, acc);
  k_final   <<<Nn, 128, 0, stream>>>(x, acc, bias, gamma, beta, out);
}